// MultiPoseLoss_85538568667826
// MI455X (gfx1250) — compile-verified
//
#include <hip/hip_runtime.h>
#include <hip/hip_bf16.h>
#include <math.h>

// ---------------------------------------------------------------------------
// CDNA5 (gfx1250) wave32 helpers
// ---------------------------------------------------------------------------
typedef __attribute__((ext_vector_type(2))) float v2f;
typedef __attribute__((ext_vector_type(8))) float v8f;

// Exact fp32 sum of one value per lane across the 32-lane wave using
// V_WMMA_F32_16X16X4_F32.
//   A (16x4 f32, 2 VGPRs): VGPR0 = v (lanes 0-15 -> K=0, lanes 16-31 -> K=2),
//                          VGPR1 = 0 (K=1 / K=3)
//   B (4x16 f32)          = all ones  -> B layout irrelevant
//   D[m,n] = v_m + v_{m+16}  (independent of n)
// Lane L sums its 8 D VGPRs:
//   lanes 0-15 : sum over m=0..7 of (v_m + v_{m+16})
//   lanes 16-31: sum over m=8..15
// One xor-16 shuffle-add completes the full wave sum on every lane.
// Requires EXEC == all ones (call only from non-divergent code).
__device__ __forceinline__ float wave_sum_f32(float v) {
    v2f a; a[0] = v;    a[1] = 0.0f;
    v2f b; b[0] = 1.0f; b[1] = 1.0f;
    v8f c = {0.f, 0.f, 0.f, 0.f, 0.f, 0.f, 0.f, 0.f};
    v8f d = __builtin_amdgcn_wmma_f32_16x16x4_f32(false, a, false, b,
                                                  (short)0, c, false, false);
    float t = ((d[0] + d[1]) + (d[2] + d[3])) + ((d[4] + d[5]) + (d[6] + d[7]));
    t += __shfl_xor(t, 16);
    return t;
}

// ---------------------------------------------------------------------------
// Problem constants
// ---------------------------------------------------------------------------
#define BB   8
#define CC   256
#define HMD  128          // heatmap side
#define PLANE (HMD * HMD) // 16384
#define OHh  11
#define OWw  8
#define FEAT (CC * OHh * OWw) // 22528
#define NRED 131072       // B * H * W per channel (and focal element count)

// ---------------------------------------------------------------------------
// Stage 1: per-(c,b) plane sum & sum-of-squares. Grid = 2048 blocks x 256.
// Each plane is 16384 contiguous floats = 4096 float4; 16 float4 per thread.
// ---------------------------------------------------------------------------
__global__ void stage1_stats(const float* __restrict__ simi,
                             float* __restrict__ ws_sum,
                             float* __restrict__ ws_ss) {
    const int cb = blockIdx.x;       // c*8 + b
    const int c = cb >> 3;
    const int b = cb & 7;
    const float4* plane =
        (const float4*)(simi + ((size_t)(b * CC + c) << 14));
    const int t = threadIdx.x;

    float s = 0.f, ss = 0.f;
#pragma unroll
    for (int i = 0; i < 16; ++i) {
        float4 x = plane[t + (i << 8)];
        s  += (x.x + x.y) + (x.z + x.w);
        ss += (x.x * x.x + x.y * x.y) + (x.z * x.z + x.w * x.w);
    }

    float wsum = wave_sum_f32(s);
    float wss  = wave_sum_f32(ss);

    __shared__ float rs[8], rss[8];
    const int lane = t & 31, wid = t >> 5;
    if (lane == 0) { rs[wid] = wsum; rss[wid] = wss; }
    __syncthreads();
    if (t == 0) {
        float S = 0.f, SS = 0.f;
#pragma unroll
        for (int w = 0; w < 8; ++w) { S += rs[w]; SS += rss[w]; }
        ws_sum[cb] = S;
        ws_ss[cb]  = SS;
    }
}

// ---------------------------------------------------------------------------
// Stage 2: finalize per-channel affine coefficients: norm(x) = a[c]*x + b[c]
// ---------------------------------------------------------------------------
__global__ void stage2_finalize(const float* __restrict__ ws_sum,
                                const float* __restrict__ ws_ss,
                                const float* __restrict__ gamma,
                                const float* __restrict__ beta,
                                float* __restrict__ ws_a,
                                float* __restrict__ ws_b) {
    const int c = threadIdx.x; // 256 threads
    float S = 0.f, SS = 0.f;
#pragma unroll
    for (int b = 0; b < 8; ++b) {
        S  += ws_sum[c * 8 + b];
        SS += ws_ss[c * 8 + b];
    }
    const float inv_n = 1.0f / (float)NRED;
    const float mean = S * inv_n;
    const float var  = SS * inv_n - mean * mean;
    const float istd = rsqrtf(var + 1e-5f);
    const float a = gamma[c] * istd;
    ws_a[c] = a;
    ws_b[c] = beta[c] - mean * a;
}

// ---------------------------------------------------------------------------
// Stage 3: triplet row sums. Grid = 16 blocks x 256 threads.
// Row i: anchor = face crop i, positive = face crop i+16, negative = mask i.
// Accumulates sum((a-p+1e-6)^2) and sum((a-n+1e-6)^2) over 22528 dims.
// ---------------------------------------------------------------------------
__global__ void stage3_triplet(const float* __restrict__ simi,
                               const int* __restrict__ bboxf,
                               const int* __restrict__ bboxm,
                               const float* __restrict__ ws_a,
                               const float* __restrict__ ws_b,
                               float* __restrict__ ws_pos,
                               float* __restrict__ ws_neg) {
    const int i = blockIdx.x;  // 0..15
    const int t = threadIdx.x;

    __shared__ int   sYa[3][OHh], sYb[3][OHh];
    __shared__ float sWy[3][OHh];
    __shared__ int   sXa[3][OWw], sXb[3][OWw];
    __shared__ float sWx[3][OWw];
    __shared__ int   sBat[3];

    if (t < 3) {
        sBat[t] = (t == 0) ? (i >> 2) : (t == 1) ? ((i >> 2) + 4) : (i >> 1);
    }
    if (t < 3 * OHh) {
        const int s = t / OHh, j = t - s * OHh;
        const int* box = (s == 0) ? (bboxf + i * 4)
                       : (s == 1) ? (bboxf + (i + 16) * 4)
                                  : (bboxm + i * 4);
        const int y0 = box[1], y1 = box[3];
        const float h = (float)(y1 - y0);
        float sy = ((float)j + 0.5f) * h / (float)OHh - 0.5f;
        sy = fminf(fmaxf(sy, 0.0f), h - 1.0f);
        const int ylo = (int)floorf(sy);
        const int yhi = min(ylo + 1, (y1 - y0) - 1);
        sWy[s][j] = sy - (float)ylo;
        sYa[s][j] = y0 + ylo;
        sYb[s][j] = y0 + yhi;
    }
    if (t >= 64 && t < 64 + 3 * OWw) {
        const int u = t - 64;
        const int s = u / OWw, k = u - s * OWw;
        const int* box = (s == 0) ? (bboxf + i * 4)
                       : (s == 1) ? (bboxf + (i + 16) * 4)
                                  : (bboxm + i * 4);
        const int x0 = box[0], x1 = box[2];
        const float w = (float)(x1 - x0);
        float sx = ((float)k + 0.5f) * w / (float)OWw - 0.5f;
        sx = fminf(fmaxf(sx, 0.0f), w - 1.0f);
        const int xlo = (int)floorf(sx);
        const int xhi = min(xlo + 1, (x1 - x0) - 1);
        sWx[s][k] = sx - (float)xlo;
        sXa[s][k] = x0 + xlo;
        sXb[s][k] = x0 + xhi;
    }
    __syncthreads();

    float accp = 0.f, accn = 0.f;
    // FEAT = 22528 = 88 * 256 -> exactly 88 iterations per thread (uniform).
    for (int d = t; d < FEAT; d += 256) {
        const int c = d / (OHh * OWw);
        const int r = d - c * (OHh * OWw);
        const int j = r >> 3, k = r & 7;
        const float ac = ws_a[c], bc = ws_b[c];
        float vals[3];
#pragma unroll
        for (int s = 0; s < 3; ++s) {
            const float* pl = simi + (((size_t)sBat[s] * CC + c) << 14);
            const int ya = sYa[s][j], yb = sYb[s][j];
            const int xa = sXa[s][k], xb = sXb[s][k];
            const float wy = sWy[s][j], wx = sWx[s][k];
            const float v00 = pl[ya * HMD + xa];
            const float v01 = pl[ya * HMD + xb];
            const float v10 = pl[yb * HMD + xa];
            const float v11 = pl[yb * HMD + xb];
            const float top = (1.f - wx) * v00 + wx * v01;
            const float bot = (1.f - wx) * v10 + wx * v11;
            const float val = (1.f - wy) * top + wy * bot;
            vals[s] = ac * val + bc;  // affine norm commutes with bilinear
        }
        const float dp = vals[0] - vals[1] + 1e-6f;
        const float dn = vals[0] - vals[2] + 1e-6f;
        accp += dp * dp;
        accn += dn * dn;
    }

    float wp = wave_sum_f32(accp);
    float wn = wave_sum_f32(accn);

    __shared__ float rp[8], rn[8];
    const int lane = t & 31, wid = t >> 5;
    if (lane == 0) { rp[wid] = wp; rn[wid] = wn; }
    __syncthreads();
    if (t == 0) {
        float P = 0.f, N = 0.f;
#pragma unroll
        for (int w = 0; w < 8; ++w) { P += rp[w]; N += rn[w]; }
        ws_pos[i] = P;
        ws_neg[i] = N;
    }
}

// ---------------------------------------------------------------------------
// Stage 4: focal-loss partials. 128 blocks x 256 threads x 4 elems = 131072.
// ---------------------------------------------------------------------------
__global__ void stage4_focal(const float* __restrict__ hm,
                             const float* __restrict__ gt,
                             float* __restrict__ ws_fl,
                             float* __restrict__ ws_fn) {
    const int t = threadIdx.x;
    const int base = blockIdx.x * 1024 + t;
    float lsum = 0.f, np = 0.f;
#pragma unroll
    for (int i = 0; i < 4; ++i) {
        const int idx = base + i * 256;
        const float x = hm[idx];
        const float g = gt[idx];
        float p = 1.0f / (1.0f + expf(-x));
        p = fminf(fmaxf(p, 1e-4f), 0.9999f);
        if (g == 1.0f) {
            const float om = 1.0f - p;
            lsum += logf(p) * om * om;
            np += 1.0f;
        } else {
            const float og = 1.0f - g;
            const float og2 = og * og;
            lsum += logf(1.0f - p) * p * p * (og2 * og2);
        }
    }

    float wl = wave_sum_f32(lsum);
    float wn = wave_sum_f32(np);

    __shared__ float rl[8], rn[8];
    const int lane = t & 31, wid = t >> 5;
    if (lane == 0) { rl[wid] = wl; rn[wid] = wn; }
    __syncthreads();
    if (t == 0) {
        float L = 0.f, N = 0.f;
#pragma unroll
        for (int w = 0; w < 8; ++w) { L += rl[w]; N += rn[w]; }
        ws_fl[blockIdx.x] = L;
        ws_fn[blockIdx.x] = N;
    }
}

// ---------------------------------------------------------------------------
// Stage 5: combine everything into the scalar loss.
// ---------------------------------------------------------------------------
__global__ void stage5_final(const float* __restrict__ ws_pos,
                             const float* __restrict__ ws_neg,
                             const float* __restrict__ ws_fl,
                             const float* __restrict__ ws_fn,
                             float* __restrict__ out) {
    if (threadIdx.x == 0) {
        float trip = 0.f;
        for (int i = 0; i < 16; ++i) {
            const float pd = sqrtf(ws_pos[i]);
            const float nd = sqrtf(ws_neg[i]);
            trip += fmaxf(0.1f + pd - nd, 0.0f);
        }
        trip *= (1.0f / 16.0f);
        float L = 0.f, N = 0.f;
        for (int w = 0; w < 128; ++w) { L += ws_fl[w]; N += ws_fn[w]; }
        const float hm_loss = -L / fmaxf(N, 1.0f);
        out[0] = hm_loss + trip;
    }
}

// ---------------------------------------------------------------------------
// Host entry
// ---------------------------------------------------------------------------
extern "C" void kernel_launch(void* const* d_in, const int* in_sizes, int n_in,
                              void* d_out, int out_size, void* d_ws, size_t ws_size,
                              hipStream_t stream) {
    const float* simi  = (const float*)d_in[0];
    const float* hm    = (const float*)d_in[1];
    const float* hm_gt = (const float*)d_in[2];
    const int*   bboxf = (const int*)d_in[3];
    const int*   bboxm = (const int*)d_in[4];
    const float* gamma = (const float*)d_in[5];
    const float* beta  = (const float*)d_in[6];
    float* out = (float*)d_out;

    float* ws      = (float*)d_ws;
    float* ws_sum  = ws;            // 2048
    float* ws_ss   = ws + 2048;     // 2048
    float* ws_a    = ws + 4096;     // 256
    float* ws_b    = ws + 4352;     // 256
    float* ws_pos  = ws + 4608;     // 16
    float* ws_neg  = ws + 4624;     // 16
    float* ws_fl   = ws + 4640;     // 128
    float* ws_fn   = ws + 4768;     // 128  (total ~19.6 KB)

    stage1_stats   <<<CC * BB, 256, 0, stream>>>(simi, ws_sum, ws_ss);
    stage2_finalize<<<1, CC, 0, stream>>>(ws_sum, ws_ss, gamma, beta, ws_a, ws_b);
    stage3_triplet <<<16, 256, 0, stream>>>(simi, bboxf, bboxm, ws_a, ws_b,
                                            ws_pos, ws_neg);
    stage4_focal   <<<128, 256, 0, stream>>>(hm, hm_gt, ws_fl, ws_fn);
    stage5_final   <<<1, 32, 0, stream>>>(ws_pos, ws_neg, ws_fl, ws_fn, out);
}